// SpatialAttention_11948599017514
// MI455X (gfx1250) — compile-verified
//
#include <hip/hip_runtime.h>

typedef __attribute__((ext_vector_type(2))) float v2f;
typedef __attribute__((ext_vector_type(8))) float v8f;
typedef __attribute__((ext_vector_type(4))) int   v4i;

#define BB 4
#define NN 512
#define RR 16
#define CC 256
#define HH 8
#define HDD 32

#define ASTR 20    // LDS stride for A tile rows (floats); 80B = 16B-aligned, conflict-free
#define BSTR 132   // LDS stride for B tile rows (floats); 528B = 16B-aligned

// ---- gfx1250 async global->LDS path (ASYNCcnt), guarded so either toolchain compiles ----
#if defined(__has_builtin)
#if __has_builtin(__builtin_amdgcn_global_load_async_to_lds_b128) && \
    __has_builtin(__builtin_amdgcn_s_wait_asynccnt)
#define USE_ASYNC_LDS 1
#endif
#endif
#ifndef USE_ASYNC_LDS
#define USE_ASYNC_LDS 0
#endif

#if USE_ASYNC_LDS
typedef __attribute__((address_space(1))) v4i glb_v4i;   // global (AS1)
typedef __attribute__((address_space(3))) v4i lds_v4i;   // LDS (AS3)
#endif

__device__ __forceinline__ void copy16_g2l(const float* __restrict__ g, float* l)
{
#if USE_ASYNC_LDS
    __builtin_amdgcn_global_load_async_to_lds_b128(
        (glb_v4i*)(void*)g, (lds_v4i*)l, 0, 0);
#else
    *(float4*)l = *(const float4*)g;
#endif
}

template <int N>
__device__ __forceinline__ void wait_async_le()
{
#if USE_ASYNC_LDS
    __builtin_amdgcn_s_wait_asynccnt(N);
#endif
}

// ---------------------------------------------------------------------------
// Pack assignment (m,n,r) -> Apack[(m*R+r), n]  (row-major, K-contiguous)
// ---------------------------------------------------------------------------
__global__ __launch_bounds__(256) void pack_assignment_kernel(
    const float* __restrict__ assign, float* __restrict__ Apack)
{
    int idx = blockIdx.x * 256 + threadIdx.x;   // 0 .. 8192*512-1
    int n   = idx & (NN - 1);
    int mr  = idx >> 9;          // / 512
    int r   = mr & (RR - 1);
    int m   = mr >> 4;
    Apack[idx] = assign[((long long)m * NN + n) * RR + r];
}

// ---------------------------------------------------------------------------
// Generic fp32 WMMA GEMM:  C[M,N] = A[M,K] * B[K,N] (+ bias[N])
// Tile: 128x128, BK=16. 256 threads = 8 waves; wave grid 4(M) x 2(N);
// each wave computes 32x64 = 2x4 tiles of v_wmma_f32_16x16x4_f32.
// Double-buffered LDS staged via GLOBAL_LOAD_ASYNC_TO_LDS_B128 (ASYNCcnt).
// Requires M%128==0, N%128==0, K%16==0 (true for all launches here).
// ---------------------------------------------------------------------------
__global__ __launch_bounds__(256) void wmma_gemm_f32_kernel(
    const float* __restrict__ A, long long sA, int lda,
    const float* __restrict__ Bm, long long sB, int ldb,
    const float* __restrict__ bias,
    float* __restrict__ Cm, long long sC, int ldc,
    int K)
{
    __shared__ float As[2][128 * ASTR];
    __shared__ float Bs[2][16 * BSTR];

    const int tid  = threadIdx.x;
    const int lane = tid & 31;
    const int w    = tid >> 5;
    const int wm   = w >> 1;        // 0..3 (M direction)
    const int wn   = w & 1;         // 0..1 (N direction)
    const int lo   = lane & 15;
    const int hi   = lane >> 4;     // 0 or 1

    const int bm = blockIdx.y * 128;
    const int bn = blockIdx.x * 128;

    const float* Ab = A  + (long long)blockIdx.z * sA;
    const float* Bb = Bm + (long long)blockIdx.z * sB;
    float*       Cb = Cm + (long long)blockIdx.z * sC;

    // Per-thread staging coordinates (4 x b128 per stage: 2 for A, 2 for B)
    const int arow0 = (tid + 0)   >> 2, ac40 = (tid + 0)   & 3;
    const int arow1 = (tid + 256) >> 2, ac41 = (tid + 256) & 3;
    const int brow0 = (tid + 0)   >> 5, bc40 = (tid + 0)   & 31;
    const int brow1 = (tid + 256) >> 5, bc41 = (tid + 256) & 31;

    v8f acc[2][4] = {};

    const int ksteps = K >> 4;

    // Issue first stage into buffer 0
    {
        copy16_g2l(Ab + (long long)(bm + arow0) * lda + 0 + ac40 * 4, &As[0][arow0 * ASTR + ac40 * 4]);
        copy16_g2l(Ab + (long long)(bm + arow1) * lda + 0 + ac41 * 4, &As[0][arow1 * ASTR + ac41 * 4]);
        copy16_g2l(Bb + (long long)(0 + brow0) * ldb + bn + bc40 * 4, &Bs[0][brow0 * BSTR + bc40 * 4]);
        copy16_g2l(Bb + (long long)(0 + brow1) * ldb + bn + bc41 * 4, &Bs[0][brow1 * BSTR + bc41 * 4]);
    }

    int buf = 0;
    for (int ki = 0; ki < ksteps; ++ki) {
        // Prefetch next stage into the other buffer while current computes
        if (ki + 1 < ksteps) {
            const int kn = (ki + 1) << 4;
            const int nb = buf ^ 1;
            copy16_g2l(Ab + (long long)(bm + arow0) * lda + kn + ac40 * 4, &As[nb][arow0 * ASTR + ac40 * 4]);
            copy16_g2l(Ab + (long long)(bm + arow1) * lda + kn + ac41 * 4, &As[nb][arow1 * ASTR + ac41 * 4]);
            copy16_g2l(Bb + (long long)(kn + brow0) * ldb + bn + bc40 * 4, &Bs[nb][brow0 * BSTR + bc40 * 4]);
            copy16_g2l(Bb + (long long)(kn + brow1) * ldb + bn + bc41 * 4, &Bs[nb][brow1 * BSTR + bc41 * 4]);
            wait_async_le<4>();   // current stage complete; only the 4 prefetches remain
        } else {
            wait_async_le<0>();   // tail: everything complete
        }
        __syncthreads();

        const float* Asb = As[buf];
        const float* Bsb = Bs[buf];
        #pragma unroll
        for (int ks = 0; ks < 4; ++ks) {
            const int kk = ks * 4 + 2 * hi;   // ISA A/B fragment layout: hi half-wave owns K+2
            v2f afrag[2];
            #pragma unroll
            for (int mt = 0; mt < 2; ++mt) {
                const float* p = &Asb[(wm * 32 + mt * 16 + lo) * ASTR + kk];
                afrag[mt] = (v2f){p[0], p[1]};
            }
            v2f bfrag[4];
            #pragma unroll
            for (int nt = 0; nt < 4; ++nt) {
                const int col = wn * 64 + nt * 16 + lo;
                bfrag[nt] = (v2f){Bsb[kk * BSTR + col], Bsb[(kk + 1) * BSTR + col]};
            }
            #pragma unroll
            for (int mt = 0; mt < 2; ++mt)
                #pragma unroll
                for (int nt = 0; nt < 4; ++nt)
                    acc[mt][nt] = __builtin_amdgcn_wmma_f32_16x16x4_f32(
                        false, afrag[mt], false, bfrag[nt],
                        (short)0, acc[mt][nt], false, false);
        }
        __syncthreads();
        buf ^= 1;
    }

    // Epilogue: D layout = lane lo -> col, VGPR j -> row j + 8*hi
    #pragma unroll
    for (int mt = 0; mt < 2; ++mt) {
        #pragma unroll
        for (int nt = 0; nt < 4; ++nt) {
            const int gc = bn + wn * 64 + nt * 16 + lo;
            const float bv = bias ? bias[gc] : 0.0f;
            #pragma unroll
            for (int j = 0; j < 8; ++j) {
                const int gr = bm + wm * 32 + mt * 16 + 8 * hi + j;
                Cb[(long long)gr * ldc + gc] = acc[mt][nt][j] + bv;
            }
        }
    }
}

// ---------------------------------------------------------------------------
// Attention: per (b,n): for each head h (one wave32, lane = d in 0..31):
//   logits[r] = <q, k_r> * scale + rel_bias[h,r]; mask -> -inf; softmax over R;
//   out[d] = sum_r p[r] * v_r[d]
// ---------------------------------------------------------------------------
__global__ __launch_bounds__(256) void attn_kernel(
    const float* __restrict__ q, const float* __restrict__ kv,
    const unsigned char* __restrict__ mask, const float* __restrict__ rel_bias,
    float* __restrict__ out)
{
    const int bn   = blockIdx.x;          // 0 .. B*N-1
    const int n    = bn & (NN - 1);
    const int h    = threadIdx.x >> 5;
    const int lane = threadIdx.x & 31;
    const float scale = 0.17677669529663687f;   // 1/sqrt(HD)

    const float qd = q[(long long)bn * CC + h * HDD + lane];

    float logits[RR];
    #pragma unroll
    for (int r = 0; r < RR; ++r) {
        const long long row = ((long long)bn * RR + r) * (2 * CC);
        float p = qd * kv[row + h * HDD + lane];
        #pragma unroll
        for (int off = 16; off > 0; off >>= 1)
            p += __shfl_xor(p, off, 32);
        p = p * scale + rel_bias[h * RR + r];
        if (mask[n * RR + r]) p = -3.0e38f;     // masked -> effectively -inf
        logits[r] = p;
    }

    float mx = logits[0];
    #pragma unroll
    for (int r = 1; r < RR; ++r) mx = fmaxf(mx, logits[r]);
    float s = 0.0f;
    #pragma unroll
    for (int r = 0; r < RR; ++r) { logits[r] = __expf(logits[r] - mx); s += logits[r]; }
    const float inv = 1.0f / s;

    float o = 0.0f;
    #pragma unroll
    for (int r = 0; r < RR; ++r) {
        const long long row = ((long long)bn * RR + r) * (2 * CC);
        o += logits[r] * inv * kv[row + CC + h * HDD + lane];
    }
    out[(long long)bn * CC + h * HDD + lane] = o;
}

// ---------------------------------------------------------------------------
// Launch
// ---------------------------------------------------------------------------
extern "C" void kernel_launch(void* const* d_in, const int* in_sizes, int n_in,
                              void* d_out, int out_size, void* d_ws, size_t ws_size,
                              hipStream_t stream)
{
    const float*         x      = (const float*)d_in[0];   // (B,N,C)
    const float*         assign = (const float*)d_in[1];   // (N,N,R)
    const unsigned char* mask   = (const unsigned char*)d_in[2]; // (N,R) bool
    const float*         Wq     = (const float*)d_in[3];   // (C,C)
    const float*         bq     = (const float*)d_in[4];   // (C,)
    const float*         Wkv    = (const float*)d_in[5];   // (C,2C)
    const float*         bkv    = (const float*)d_in[6];   // (2C,)
    const float*         rel    = (const float*)d_in[7];   // (H,1,R)
    const float*         Wp     = (const float*)d_in[8];   // (C,C)
    const float*         bp     = (const float*)d_in[9];   // (C,)
    float*               outp   = (float*)d_out;           // (B,N,C)

    float* ws    = (float*)d_ws;
    float* Apack = ws;                                   // 8192*512
    float* prekv = Apack + (size_t)8192 * 512;           // 32768*256
    float* kvbuf = prekv + (size_t)32768 * 256;          // 32768*512
    float* qbuf  = kvbuf + (size_t)32768 * 512;          // 2048*256
    float* abuf  = qbuf  + (size_t)2048 * 256;           // 2048*256

    // 1) Pack assignment into GEMM-A layout
    pack_assignment_kernel<<<(8192 * 512) / 256, 256, 0, stream>>>(assign, Apack);

    // 2) pre_kv[b] (8192x256) = Apack (8192x512) @ x_b (512x256), batch=B
    wmma_gemm_f32_kernel<<<dim3(256 / 128, 8192 / 128, BB), 256, 0, stream>>>(
        Apack, 0LL, 512,
        x, (long long)NN * CC, CC,
        nullptr,
        prekv, (long long)8192 * CC, CC,
        512);

    // 3) kv (32768x512) = pre_kv (32768x256) @ Wkv (256x512) + bkv
    wmma_gemm_f32_kernel<<<dim3(512 / 128, 32768 / 128, 1), 256, 0, stream>>>(
        prekv, 0LL, 256,
        Wkv, 0LL, 512,
        bkv,
        kvbuf, 0LL, 512,
        256);

    // 4) q (2048x256) = x @ Wq + bq
    wmma_gemm_f32_kernel<<<dim3(256 / 128, 2048 / 128, 1), 256, 0, stream>>>(
        x, 0LL, 256,
        Wq, 0LL, 256,
        bq,
        qbuf, 0LL, 256,
        256);

    // 5) attention (softmax over R=16 per (b,n,h))
    attn_kernel<<<BB * NN, 256, 0, stream>>>(qbuf, kvbuf, mask, rel, abuf);

    // 6) out (2048x256) = attn_out @ Wp + bp
    wmma_gemm_f32_kernel<<<dim3(256 / 128, 2048 / 128, 1), 256, 0, stream>>>(
        abuf, 0LL, 256,
        Wp, 0LL, 256,
        bp,
        outp, 0LL, 256,
        256);
}